// HGTBackbone_32770600468608
// MI455X (gfx1250) — compile-verified
//
#include <hip/hip_runtime.h>
#include <math.h>

typedef float v2f __attribute__((ext_vector_type(2)));
typedef float v8f __attribute__((ext_vector_type(8)));

__device__ __forceinline__ float gelu_f(float x) {
    return 0.5f * x * (1.0f + erff(x * 0.70710678118654752440f));
}
// order-preserving float<->uint map for atomicMax on floats
__device__ __forceinline__ unsigned ord_map(float f) {
    unsigned u = __float_as_uint(f);
    return (u >> 31) ? ~u : (u | 0x80000000u);
}
__device__ __forceinline__ float ord_unmap(unsigned m) {
    return (m >> 31) ? __uint_as_float(m & 0x7fffffffu) : __uint_as_float(~m);
}

// ---------------------------------------------------------------------------
// Y[n,128] = act(X)[n,128] @ W[128,128] + bias, via V_WMMA_F32_16X16X4_F32.
// Block: 256 threads = 8 waves; wave w computes rows [blk*128+w*16, +16) x 128.
// A rows are clamped (not predicated): garbage rows only feed unstored C rows.
// W staged in LDS pair-interleaved: Wl[p*128+col] = {W[2p][col], W[2p+1][col]}
// so each B fragment is one contiguous ds_load_b64 into an even VGPR pair.
// ---------------------------------------------------------------------------
template <bool GELU>
__global__ __launch_bounds__(256) void hgt_gemm128(
    const float* __restrict__ X, const float* __restrict__ W,
    const float* __restrict__ bias, float* __restrict__ Y, int nrows)
{
    __shared__ float2 Wl[64 * 128];   // 64 KB, pair-interleaved
    const int tid = threadIdx.x;
    for (int i = tid; i < 64 * 128; i += 256) {
        const int p = i >> 7, col = i & 127;
        float2 w;
        w.x = W[(size_t)(2 * p) * 128 + col];
        w.y = W[(size_t)(2 * p + 1) * 128 + col];
        Wl[i] = w;
    }
    __syncthreads();

    const int wave = tid >> 5;
    const int lane = tid & 31;
    const int half = lane >> 4;      // 0: lanes 0-15, 1: lanes 16-31
    const int l16  = lane & 15;
    const int rowBase = blockIdx.x * 128 + wave * 16;
    int row = rowBase + l16;         // A-matrix: M = lane%16 for both halves
    if (row >= nrows) row = nrows - 1;   // clamp: unconditional loads
    const float2* xr2 = (const float2*)(X + (size_t)row * 128);
    __builtin_prefetch(xr2, 0, 3);   // global_prefetch_b8

    v8f c[8];
    #pragma unroll
    for (int t = 0; t < 8; ++t)
        #pragma unroll
        for (int r = 0; r < 8; ++r) c[t][r] = 0.0f;

    #pragma unroll 4
    for (int k0 = 0; k0 < 128; k0 += 4) {
        // A frag: VGPR0 holds K = k0 + (half?2:0), VGPR1 holds K+1 (ISA 7.12.2)
        const int kp = (k0 >> 1) + half;       // K-pair index (ka>>1)
        float2 a01 = xr2[kp];                  // one global_load_b64
        if (GELU) { a01.x = gelu_f(a01.x); a01.y = gelu_f(a01.y); }
        v2f a; a.x = a01.x; a.y = a01.y;
        #pragma unroll
        for (int t = 0; t < 8; ++t) {
            const int col = t * 16 + l16;
            const float2 bw = Wl[kp * 128 + col];   // one ds_load_b64
            v2f b; b.x = bw.x; b.y = bw.y;
            c[t] = __builtin_amdgcn_wmma_f32_16x16x4_f32(false, a, false, b,
                                                         (short)0, c[t], false, false);
        }
    }

    #pragma unroll
    for (int t = 0; t < 8; ++t) {
        const int col = t * 16 + l16;
        const float bv = bias[col];
        #pragma unroll
        for (int r = 0; r < 8; ++r) {   // C/D: VGPR r -> M=r (lo half) / r+8 (hi half)
            const int rr = rowBase + (half ? r + 8 : r);
            if (rr < nrows) Y[(size_t)rr * 128 + col] = c[t][r] + bv;
        }
    }
}

// ---------------------------------------------------------------------------
// Edge pass 1: logit[e,h] = (q_dst . (k_src . rel_a[h] + ea)) * p[h]/sqrt(32)
// where ea[f] = be[h*32+f] + sum_j et[e,j]*We[j, h*32+f]. Also segment max.
// One thread per (edge, head).
// ---------------------------------------------------------------------------
__global__ __launch_bounds__(256) void hgt_edge_logits(
    const float* __restrict__ K, const float* __restrict__ Q,
    const int* __restrict__ src, const int* __restrict__ dst,
    const float* __restrict__ et, const float* __restrict__ We,
    const float* __restrict__ be, const float* __restrict__ ra,
    const float* __restrict__ pr, float* __restrict__ lg,
    unsigned* __restrict__ nmax, int nE)
{
    __shared__ float sWe[9 * 128];
    __shared__ float sbe[128];
    __shared__ float sra[4 * 32 * 32];
    __shared__ float sp[4];
    const int tid = threadIdx.x;
    for (int i = tid; i < 9 * 128; i += 256) sWe[i] = We[i];
    for (int i = tid; i < 128; i += 256) sbe[i] = be[i];
    for (int i = tid; i < 4096; i += 256) sra[i] = ra[i];
    if (tid < 4) sp[tid] = pr[tid];
    __syncthreads();

    const int g = blockIdx.x * 256 + tid;
    if (g >= nE * 4) return;
    const int e = g >> 2, h = g & 3;
    const int s = src[e], d = dst[e];
    const float* kv = K + (size_t)s * 128 + h * 32;
    const float* qv = Q + (size_t)d * 128 + h * 32;

    float ej[9];
    #pragma unroll
    for (int j = 0; j < 9; ++j) ej[j] = et[(size_t)e * 9 + j];
    float q[32];
    #pragma unroll
    for (int f = 0; f < 32; ++f) q[f] = qv[f];

    // acc = q . ea
    float acc = 0.0f;
    for (int f = 0; f < 32; ++f) {
        float ea = sbe[h * 32 + f];
        #pragma unroll
        for (int j = 0; j < 9; ++j) ea += ej[j] * sWe[j * 128 + h * 32 + f];
        acc += q[f] * ea;
    }
    // acc += sum_d k[d] * (rel_a[h][d,:] . q)
    const float* A = sra + h * 1024;
    for (int d0 = 0; d0 < 32; ++d0) {
        float t = 0.0f;
        #pragma unroll
        for (int f = 0; f < 32; ++f) t += A[d0 * 32 + f] * q[f];
        acc += kv[d0] * t;
    }
    const float logit = acc * sp[h] * 0.17677669529663688f;  // 1/sqrt(32)
    lg[g] = logit;
    atomicMax(&nmax[(size_t)d * 4 + h], ord_map(logit));
}

// ---------------------------------------------------------------------------
// Edge pass 2: numerator exp(logit - max) (stored in-place) + segment sum.
// ---------------------------------------------------------------------------
__global__ __launch_bounds__(256) void hgt_edge_alpha(
    const int* __restrict__ dst, float* __restrict__ lg,
    const unsigned* __restrict__ nmax, float* __restrict__ denom, int nE)
{
    const int g = blockIdx.x * 256 + threadIdx.x;
    if (g >= nE * 4) return;
    const int e = g >> 2, h = g & 3;
    const int d = dst[e];
    const float m = ord_unmap(nmax[(size_t)d * 4 + h]);
    const float ex = expf(lg[g] - m);
    lg[g] = ex;
    atomicAdd(&denom[(size_t)d * 4 + h], ex);
}

// ---------------------------------------------------------------------------
// Edge pass 3: ve = v_src . rel_m[h] + ea; agg[dst] += ve * alpha/(den+eps)
// ---------------------------------------------------------------------------
__global__ __launch_bounds__(256) void hgt_edge_agg(
    const float* __restrict__ V, const int* __restrict__ src,
    const int* __restrict__ dst, const float* __restrict__ et,
    const float* __restrict__ We, const float* __restrict__ be,
    const float* __restrict__ rm, const float* __restrict__ lg,
    const float* __restrict__ denom, float* __restrict__ agg, int nE)
{
    __shared__ float sWe[9 * 128];
    __shared__ float sbe[128];
    __shared__ float srm[4 * 32 * 32];
    const int tid = threadIdx.x;
    for (int i = tid; i < 9 * 128; i += 256) sWe[i] = We[i];
    for (int i = tid; i < 128; i += 256) sbe[i] = be[i];
    for (int i = tid; i < 4096; i += 256) srm[i] = rm[i];
    __syncthreads();

    const int g = blockIdx.x * 256 + tid;
    if (g >= nE * 4) return;
    const int e = g >> 2, h = g & 3;
    const int s = src[e], d = dst[e];
    const float w = lg[g] / (denom[(size_t)d * 4 + h] + 1e-16f);

    float ej[9];
    #pragma unroll
    for (int j = 0; j < 9; ++j) ej[j] = et[(size_t)e * 9 + j];
    float v[32];
    const float* vv = V + (size_t)s * 128 + h * 32;
    #pragma unroll
    for (int d0 = 0; d0 < 32; ++d0) v[d0] = vv[d0];

    const float* M = srm + h * 1024;
    float* ao = agg + (size_t)d * 128 + h * 32;
    for (int f = 0; f < 32; ++f) {
        float ve = sbe[h * 32 + f];
        #pragma unroll
        for (int j = 0; j < 9; ++j) ve += ej[j] * sWe[j * 128 + h * 32 + f];
        #pragma unroll
        for (int d0 = 0; d0 < 32; ++d0) ve += v[d0] * M[d0 * 32 + f];
        atomicAdd(&ao[f], ve * w);
    }
}

// ---------------------------------------------------------------------------
// out = LayerNorm(beta*o + (1-beta)*hprev) * g + b ; one wave per row.
// (o/hprev/out may alias elementwise -> no __restrict__ here)
// ---------------------------------------------------------------------------
__global__ __launch_bounds__(256) void hgt_gate_norm(
    const float* o, const float* hprev, const float* skip,
    const float* gam, const float* bet, float* out, int nrows)
{
    const int wave = threadIdx.x >> 5, lane = threadIdx.x & 31;
    const int row = blockIdx.x * 8 + wave;
    if (row >= nrows) return;
    const float beta = 1.0f / (1.0f + expf(-skip[0]));
    const size_t base = (size_t)row * 128;
    float y[4];
    float s = 0.0f;
    #pragma unroll
    for (int i = 0; i < 4; ++i) {
        const int c = lane + i * 32;
        const float v = beta * o[base + c] + (1.0f - beta) * hprev[base + c];
        y[i] = v; s += v;
    }
    #pragma unroll
    for (int off = 16; off > 0; off >>= 1) s += __shfl_xor(s, off);
    const float mu = s * (1.0f / 128.0f);
    float vs = 0.0f;
    #pragma unroll
    for (int i = 0; i < 4; ++i) { const float d = y[i] - mu; vs += d * d; }
    #pragma unroll
    for (int off = 16; off > 0; off >>= 1) vs += __shfl_xor(vs, off);
    const float inv = rsqrtf(vs * (1.0f / 128.0f) + 1e-5f);
    #pragma unroll
    for (int i = 0; i < 4; ++i) {
        const int c = lane + i * 32;
        out[base + c] = (y[i] - mu) * inv * gam[c] + bet[c];
    }
}

// ---------------------------------------------------------------------------
// Host driver. Input flattening = jax pytree order (dict keys sorted):
// 0 x_a, 1 x_b, 2 ei_ab, 3 ei_ba, 4 et_ab, 5 et_ba,
// 6 elin.ab.b, 7 elin.ab.w, 8 elin.ba.b, 9 elin.ba.w,
// per layer l (base 10+24l): 0 a.A.b 1 a.A.w 2 a.B.b 3 a.B.w 4 k.A.b 5 k.A.w
//   6 k.B.b 7 k.B.w 8 q.A.b 9 q.A.w 10 q.B.b 11 q.B.w 12 rel.ab.a 13 rel.ab.m
//   14 rel.ab.p 15 rel.ba.a 16 rel.ba.m 17 rel.ba.p 18 skip.A 19 skip.B
//   20 v.A.b 21 v.A.w 22 v.B.b 23 v.B.w
// 82 norm.A.b, 83 norm.A.g, 84 norm.B.b, 85 norm.B.g
// ---------------------------------------------------------------------------
extern "C" void kernel_launch(void* const* d_in, const int* in_sizes, int n_in,
                              void* d_out, int out_size, void* d_ws, size_t ws_size,
                              hipStream_t stream)
{
    const float* x_a  = (const float*)d_in[0];
    const float* x_b  = (const float*)d_in[1];
    const int*  ei_ab = (const int*)d_in[2];
    const int*  ei_ba = (const int*)d_in[3];
    const float* et_ab = (const float*)d_in[4];
    const float* et_ba = (const float*)d_in[5];
    const float* elb_ab = (const float*)d_in[6];
    const float* elw_ab = (const float*)d_in[7];
    const float* elb_ba = (const float*)d_in[8];
    const float* elw_ba = (const float*)d_in[9];

    const int N = in_sizes[0] / 128;
    const int E = in_sizes[2] / 2;

    const int normBase = 10 + 24 * 3;
    const float* nA_b = (const float*)d_in[normBase + 0];
    const float* nA_g = (const float*)d_in[normBase + 1];
    const float* nB_b = (const float*)d_in[normBase + 2];
    const float* nB_g = (const float*)d_in[normBase + 3];

    const size_t ND = (size_t)N * 128;
    float* f = (float*)d_ws;
    float* kA = f;           float* kB = f + ND;
    float* qA = f + 2 * ND;  float* qB = f + 3 * ND;
    float* vA = f + 4 * ND;  float* vB = f + 5 * ND;
    float* aggA = f + 6 * ND; float* aggB = f + 7 * ND;
    float* hA = f + 8 * ND;  float* hB = f + 9 * ND;
    float* lg = f + 10 * ND;                           // E*4 floats
    unsigned* nmax = (unsigned*)(lg + (size_t)E * 4);  // N*4 uints
    float* denom = (float*)(nmax + (size_t)N * 4);     // N*4 floats

    const int gemmBlocks = (N + 127) / 128;
    const int edgeBlocks = (E * 4 + 255) / 256;
    const int rowBlocks  = (N + 7) / 8;

    const float* curA = x_a;
    const float* curB = x_b;
    float* outA = (float*)d_out;
    float* outB = (float*)d_out + ND;

    for (int l = 0; l < 3; ++l) {
        const int b0 = 10 + 24 * l;
        auto P = [&](int i) { return (const float*)d_in[b0 + i]; };

        // K/Q/V projections (WMMA GEMMs)
        hgt_gemm128<false><<<gemmBlocks, 256, 0, stream>>>(curA, P(5),  P(4),  kA, N);
        hgt_gemm128<false><<<gemmBlocks, 256, 0, stream>>>(curB, P(7),  P(6),  kB, N);
        hgt_gemm128<false><<<gemmBlocks, 256, 0, stream>>>(curA, P(9),  P(8),  qA, N);
        hgt_gemm128<false><<<gemmBlocks, 256, 0, stream>>>(curB, P(11), P(10), qB, N);
        hgt_gemm128<false><<<gemmBlocks, 256, 0, stream>>>(curA, P(21), P(20), vA, N);
        hgt_gemm128<false><<<gemmBlocks, 256, 0, stream>>>(curB, P(23), P(22), vB, N);

        hipMemsetAsync(aggA, 0, ND * sizeof(float), stream);
        hipMemsetAsync(aggB, 0, ND * sizeof(float), stream);

        // edge type "ab": src A -> dst B
        hipMemsetAsync(nmax, 0, (size_t)N * 4 * sizeof(unsigned), stream);
        hipMemsetAsync(denom, 0, (size_t)N * 4 * sizeof(float), stream);
        hgt_edge_logits<<<edgeBlocks, 256, 0, stream>>>(
            kA, qB, ei_ab, ei_ab + E, et_ab, elw_ab, elb_ab, P(12), P(14), lg, nmax, E);
        hgt_edge_alpha<<<edgeBlocks, 256, 0, stream>>>(ei_ab + E, lg, nmax, denom, E);
        hgt_edge_agg<<<edgeBlocks, 256, 0, stream>>>(
            vA, ei_ab, ei_ab + E, et_ab, elw_ab, elb_ab, P(13), lg, denom, aggB, E);

        // edge type "ba": src B -> dst A
        hipMemsetAsync(nmax, 0, (size_t)N * 4 * sizeof(unsigned), stream);
        hipMemsetAsync(denom, 0, (size_t)N * 4 * sizeof(float), stream);
        hgt_edge_logits<<<edgeBlocks, 256, 0, stream>>>(
            kB, qA, ei_ba, ei_ba + E, et_ba, elw_ba, elb_ba, P(15), P(17), lg, nmax, E);
        hgt_edge_alpha<<<edgeBlocks, 256, 0, stream>>>(ei_ba + E, lg, nmax, denom, E);
        hgt_edge_agg<<<edgeBlocks, 256, 0, stream>>>(
            vB, ei_ba, ei_ba + E, et_ba, elw_ba, elb_ba, P(16), lg, denom, aggA, E);

        // output projection with exact GELU on agg (reuse kA/kB as o buffers)
        hgt_gemm128<true><<<gemmBlocks, 256, 0, stream>>>(aggA, P(1), P(0), kA, N);
        hgt_gemm128<true><<<gemmBlocks, 256, 0, stream>>>(aggB, P(3), P(2), kB, N);

        float* dstA = (l == 2) ? outA : hA;
        float* dstB = (l == 2) ? outB : hB;
        hgt_gate_norm<<<rowBlocks, 256, 0, stream>>>(kA, curA, P(18), nA_g, nA_b, dstA, N);
        hgt_gate_norm<<<rowBlocks, 256, 0, stream>>>(kB, curB, P(19), nB_g, nB_b, dstB, N);
        curA = dstA;
        curB = dstB;
    }
}